// MoERouter_15496242004073
// MI455X (gfx1250) — compile-verified
//
#include <hip/hip_runtime.h>
#include <hip/hip_bf16.h>
#include <math.h>

typedef __attribute__((ext_vector_type(2))) float v2f;
typedef __attribute__((ext_vector_type(8))) float v8f;

#define D_DIM 2048
#define E_DIM 64
#define TOK_PER_BLK 64
#define KCHUNK 128
#define ZLOSS_COEF 0.001f

// One block = 4 waves (wave32) = 64 tokens. Each wave owns a 16-token x 64-expert
// logits tile held in 4 v8f WMMA accumulators (V_WMMA_F32_16X16X4_F32, fp32 path).
// W is staged through LDS in 128-row chunks, stored K-PAIR INTERLEAVED
// ( lds[(p*64+n)*2 + j] = W[2p+j][n] ) so every B fragment is a single aligned
// ds_load_b64 straight into an adjacent VGPR pair — no repack movs before WMMA.
// The x stream (134 MB, the HBM-bound stream) is prefetched one full chunk ahead,
// 4 cachelines per lane, to put ~16 MB in flight chip-wide (BW x latency product).
__global__ __launch_bounds__(128) void router_fused_kernel(
    const float* __restrict__ x,      // [nTok, 2048]
    const float* __restrict__ W,      // [2048, 64]
    const float* __restrict__ bias,   // [64]
    float* __restrict__ out,          // probs[2*nTok] | idx[2*nTok] | zloss[1]
    float* __restrict__ lse2,         // [nTok] scratch
    int nTok) {
  __shared__ float lds[KCHUNK * E_DIM];  // 8192 floats (32 KB), reused for logits

  const int tid  = threadIdx.x;
  const int wave = tid >> 5;
  const int lane = tid & 31;
  const int l16  = lane & 15;
  const int kh   = lane >> 4;   // 0: K pair {0,1}, 1: K pair {2,3}

  const int rowBase = blockIdx.x * TOK_PER_BLK + wave * 16;
  const float* __restrict__ xrow = x + (size_t)(rowBase + l16) * D_DIM;

  v8f acc0 = {}, acc1 = {}, acc2 = {}, acc3 = {};

  // Prefetch the first chunk of this lane's x row (4 x 128B lines).
#pragma unroll
  for (int pl = 0; pl < 4; ++pl)
    __builtin_prefetch(xrow + pl * 32, 0, 0);

  for (int k0 = 0; k0 < D_DIM; k0 += KCHUNK) {
    __syncthreads();
    // Cooperative load of W[k0..k0+127][0..63] -> LDS, k-pair interleaved.
    // 64 row-pairs x 16 col-quads = 1024 items; 8 items/thread.
    // Global side: 16 consecutive tids stream 256B runs of each row (coalesced).
    {
#pragma unroll
      for (int j = 0; j < 8; ++j) {
        const int item = tid + 128 * j;
        const int p  = item >> 4;        // row pair 0..63
        const int cq = item & 15;        // col quad 0..15
        const float4 ra = *(const float4*)(W + (size_t)(k0 + 2 * p)     * E_DIM + cq * 4);
        const float4 rb = *(const float4*)(W + (size_t)(k0 + 2 * p + 1) * E_DIM + cq * 4);
        v2f* dst = (v2f*)(lds + (p * E_DIM + cq * 4) * 2);
        dst[0] = v2f{ra.x, rb.x};
        dst[1] = v2f{ra.y, rb.y};
        dst[2] = v2f{ra.z, rb.z};
        dst[3] = v2f{ra.w, rb.w};
      }
    }
    __syncthreads();

    // Prefetch the ENTIRE next chunk of this lane's x row (512B = 4 lines),
    // overlapped with this chunk's WMMA work.
    if (k0 + KCHUNK < D_DIM) {
#pragma unroll
      for (int pl = 0; pl < 4; ++pl)
        __builtin_prefetch(xrow + k0 + KCHUNK + pl * 32, 0, 0);
    }

#pragma unroll 4
    for (int kk = 0; kk < KCHUNK; kk += 4) {
      // A fragment: lane (l16, kh) loads x[row=l16][k0+kk+2*kh .. +1] as b64.
      v2f a = *(const v2f*)(xrow + k0 + kk + kh * 2);

      // B fragments: k-pair p = kk/2 + kh, each fragment one aligned b64.
      const float* __restrict__ bbase = lds + ((kk / 2 + kh) * E_DIM + l16) * 2;
      v2f b0 = *(const v2f*)(bbase + 0);
      v2f b1 = *(const v2f*)(bbase + 32);
      v2f b2 = *(const v2f*)(bbase + 64);
      v2f b3 = *(const v2f*)(bbase + 96);

      acc0 = __builtin_amdgcn_wmma_f32_16x16x4_f32(false, a, false, b0, (short)0, acc0, false, false);
      acc1 = __builtin_amdgcn_wmma_f32_16x16x4_f32(false, a, false, b1, (short)0, acc1, false, false);
      acc2 = __builtin_amdgcn_wmma_f32_16x16x4_f32(false, a, false, b2, (short)0, acc2, false, false);
      acc3 = __builtin_amdgcn_wmma_f32_16x16x4_f32(false, a, false, b3, (short)0, acc3, false, false);
    }
  }

  // Bias: C/D layout puts N in lanes, so bias is a per-lane scalar per N-tile.
  {
    const float bb0 = bias[l16], bb1 = bias[16 + l16],
                bb2 = bias[32 + l16], bb3 = bias[48 + l16];
#pragma unroll
    for (int r = 0; r < 8; ++r) {
      acc0[r] += bb0; acc1[r] += bb1; acc2[r] += bb2; acc3[r] += bb3;
    }
  }

  __syncthreads();
  // Scatter logits to LDS, row stride 65 (bank-conflict-free token reads).
  // C/D layout: VGPR r holds M = r + 8*kh (lanes 0-15 -> M 0..7, lanes 16-31 -> M 8..15).
#pragma unroll
  for (int r = 0; r < 8; ++r) {
    const int m = wave * 16 + kh * 8 + r;
    lds[m * 65 + l16]      = acc0[r];
    lds[m * 65 + 16 + l16] = acc1[r];
    lds[m * 65 + 32 + l16] = acc2[r];
    lds[m * 65 + 48 + l16] = acc3[r];
  }
  __syncthreads();

  // Per-token softmax statistics + top-2 (+ renorm) + lse^2, one thread/token.
  if (tid < TOK_PER_BLK) {
    const int token = blockIdx.x * TOK_PER_BLK + tid;
    const float* __restrict__ lrow = lds + tid * 65;

    float mx = -INFINITY;
    float m1 = -INFINITY, m2 = -INFINITY;
    int i1 = 0, i2 = 0;
    for (int e = 0; e < E_DIM; ++e) {
      const float v = lrow[e];
      mx = fmaxf(mx, v);
      if (v > m1) { m2 = m1; i2 = i1; m1 = v; i1 = e; }
      else if (v > m2) { m2 = v; i2 = e; }
    }
    float s = 0.f;
    for (int e = 0; e < E_DIM; ++e) s += expf(lrow[e] - mx);

    const float lse = mx + logf(s);
    lse2[token] = lse * lse;

    const float e1 = expf(m1 - mx);
    const float e2 = expf(m2 - mx);
    const float inv = 1.0f / (e1 + e2);

    const int N2 = nTok * 2;
    out[token * 2 + 0]      = e1 * inv;
    out[token * 2 + 1]      = e2 * inv;
    out[N2 + token * 2 + 0] = (float)i1;
    out[N2 + token * 2 + 1] = (float)i2;
  }
}

// Deterministic single-block reduction of lse^2 -> z-loss scalar.
__global__ __launch_bounds__(256) void zloss_reduce_kernel(
    const float* __restrict__ lse2, float* __restrict__ out, int nTok) {
  __shared__ float part[256];
  float s = 0.f;
  for (int i = threadIdx.x; i < nTok; i += 256) s += lse2[i];
  part[threadIdx.x] = s;
  __syncthreads();
  for (int off = 128; off > 0; off >>= 1) {
    if ((int)threadIdx.x < off) part[threadIdx.x] += part[threadIdx.x + off];
    __syncthreads();
  }
  if (threadIdx.x == 0)
    out[4 * nTok] = (part[0] / (float)nTok) * ZLOSS_COEF;
}

extern "C" void kernel_launch(void* const* d_in, const int* in_sizes, int n_in,
                              void* d_out, int out_size, void* d_ws, size_t ws_size,
                              hipStream_t stream) {
  const float* x    = (const float*)d_in[0];  // [B*S, D] = [16384, 2048]
  const float* W    = (const float*)d_in[1];  // [2048, 64]
  const float* bias = (const float*)d_in[2];  // [64]
  float* out  = (float*)d_out;
  float* lse2 = (float*)d_ws;

  const int nTok = in_sizes[0] / D_DIM;       // 16384
  const int nBlocks = nTok / TOK_PER_BLK;     // 256

  router_fused_kernel<<<nBlocks, 128, 0, stream>>>(x, W, bias, out, lse2, nTok);
  zloss_reduce_kernel<<<1, 256, 0, stream>>>(lse2, out, nTok);
}